// Mamba_77257871720650
// MI455X (gfx1250) — compile-verified
//
#include <hip/hip_runtime.h>
#include <hip/hip_bf16.h>

// Mamba (2-layer) forward for MI455X / gfx1250, fp32 end-to-end.
// GEMMs: V_WMMA_F32_16X16X4_F32, 128x128 block tiles, double-buffered LDS
// staging via gfx1250 GLOBAL_LOAD_ASYNC_TO_LDS_B128 (ASYNCcnt-tracked).
// Selective scan: 3-phase segmented linear scan (8x parallelism over L).

typedef float v2f __attribute__((ext_vector_type(2)));
typedef float v8f __attribute__((ext_vector_type(8)));
typedef int   v4i __attribute__((ext_vector_type(4)));

#define ROWS 2048   // B*L
#define DM   768
#define DI   1536
#define VOC  50264
#define SEG    8
#define SEGLEN 128  // 1024/SEG

#if defined(__has_builtin)
#if __has_builtin(__builtin_amdgcn_global_load_async_to_lds_b128) && \
    __has_builtin(__builtin_amdgcn_s_wait_asynccnt)
#define ASYNC_OK 1
#endif
#endif
#ifndef ASYNC_OK
#define ASYNC_OK 0
#endif

#if ASYNC_OK
typedef __attribute__((address_space(1))) v4i* gptr_v4i;
typedef __attribute__((address_space(3))) v4i* lptr_v4i;
__device__ __forceinline__ void async_cp16(const float* g, float* l){
  __builtin_amdgcn_global_load_async_to_lds_b128(
      (gptr_v4i)(void*)g, (lptr_v4i)(void*)l, 0, 0);
}
#endif

__device__ __forceinline__ float sigmoidf_(float x){ return 1.f/(1.f+__expf(-x)); }
__device__ __forceinline__ float softplusf_(float x){ return x>20.f ? x : log1pf(__expf(x)); }

// ---------------- embedding gather ----------------
__global__ void embed_kernel(const int* __restrict__ ids, const float* __restrict__ emb,
                             float* __restrict__ x)
{
  const int row = blockIdx.x;
  const int id  = ids[row];
  const float* src = emb + (size_t)id*DM;
  float* dst = x + (size_t)row*DM;
  for (int c = threadIdx.x; c < DM; c += blockDim.x) dst[c] = src[c];
}

// ---------------- rmsnorm (one block per row) ----------------
__global__ void rmsnorm_kernel(const float* __restrict__ x, const float* __restrict__ w,
                               float* __restrict__ out)
{
  __shared__ float red[256];
  const int row = blockIdx.x;
  const float* xr = x + (size_t)row*DM;
  float s = 0.f;
  for (int c = threadIdx.x; c < DM; c += 256){ float v = xr[c]; s += v*v; }
  red[threadIdx.x] = s; __syncthreads();
  for (int off = 128; off > 0; off >>= 1){
    if (threadIdx.x < off) red[threadIdx.x] += red[threadIdx.x+off];
    __syncthreads();
  }
  const float scale = rsqrtf(red[0]*(1.f/(float)DM) + 1e-5f);
  float* o = out + (size_t)row*DM;
  for (int c = threadIdx.x; c < DM; c += 256) o[c] = xr[c]*scale*w[c];
}

// ---------------- tiled WMMA GEMM: C[M,N] = act(A[M,K] @ W[N,K]^T + bias) (+Res) ----
// 256 threads = 8 waves; block tile 128(M) x 128(N); wave tile 32(M) x 64(N).
// K staged in KC=16 chunks, double-buffered in LDS (row stride 20 floats:
// conflict-free b64 fragment reads, 16B-aligned async b128 stores).
// grid = (M/128, ceil(N/128));  ACT: 0=none, 1=softplus;  NT: streaming stores
template<int ACT, bool HB, bool HR, bool NT>
__global__ __launch_bounds__(256) void gemm_tiled(
    const float* __restrict__ A, int lda,
    const float* __restrict__ W, int ldw,
    const float* __restrict__ bias,
    const float* __restrict__ Res,
    float* __restrict__ C,
    int M, int N, int K)
{
  constexpr int KC = 16, LDSW = 20;
  __shared__ float As[2][128*LDSW];
  __shared__ float Bs[2][128*LDSW];

  const int tid  = threadIdx.x;
  const int lane = tid & 31, wave = tid >> 5;
  const int wm   = wave & 3;        // wave M index (0..3) -> 32 rows each
  const int wn   = wave >> 2;       // wave N index (0..1) -> 64 cols each
  const int half = lane >> 4, l16 = lane & 15;
  const int mBase = blockIdx.x * 128;
  const int nBase = blockIdx.y * 128;

  // ---- staging addresses: thread -> (row sr / sr+64, 4 consecutive K) ----
  const int sr = tid >> 2;          // 0..63
  const int sp = (tid & 3) * 4;     // 0,4,8,12 within chunk
  const float* aG0 = A + (size_t)(mBase + sr)      * lda + sp;
  const float* aG1 = A + (size_t)(mBase + sr + 64) * lda + sp;
  int wr0 = nBase + sr;      if (wr0 >= N) wr0 = N - 1;   // clamp; masked at store
  int wr1 = nBase + sr + 64; if (wr1 >= N) wr1 = N - 1;
  const float* bG0 = W + (size_t)wr0 * ldw + sp;
  const float* bG1 = W + (size_t)wr1 * ldw + sp;
  const int soff0 = sr * LDSW + sp;
  const int soff1 = (sr + 64) * LDSW + sp;

  // ---- fragment LDS offsets ----
  const int aOff0 = (wm*32 +  0 + l16) * LDSW + half*2;
  const int aOff1 = (wm*32 + 16 + l16) * LDSW + half*2;
  int bOff[4];
  #pragma unroll
  for (int j = 0; j < 4; j++) bOff[j] = (wn*64 + j*16 + l16) * LDSW + half*2;

  v8f acc[2][4];
  #pragma unroll
  for (int g = 0; g < 2; g++)
    #pragma unroll
    for (int j = 0; j < 4; j++)
      #pragma unroll
      for (int v = 0; v < 8; v++) acc[g][j][v] = 0.f;

  const int nCh = K / KC;

  // ---- prologue: stage chunk 0 into buffer 0 ----
#if ASYNC_OK
  async_cp16(aG0, &As[0][soff0]);
  async_cp16(aG1, &As[0][soff1]);
  async_cp16(bG0, &Bs[0][soff0]);
  async_cp16(bG1, &Bs[0][soff1]);
  __builtin_amdgcn_s_wait_asynccnt(0);
#else
  {
    float4 ra0 = *(const float4*)aG0, ra1 = *(const float4*)aG1;
    float4 rb0 = *(const float4*)bG0, rb1 = *(const float4*)bG1;
    *(float4*)&As[0][soff0] = ra0;  *(float4*)&As[0][soff1] = ra1;
    *(float4*)&Bs[0][soff0] = rb0;  *(float4*)&Bs[0][soff1] = rb1;
  }
#endif
  __syncthreads();

  for (int c = 0; c < nCh; c++){
    const int cur = c & 1, nxt = cur ^ 1;
    const bool more = (c + 1 < nCh);
    const int kN = (c + 1) * KC;

#if ASYNC_OK
    if (more){
      async_cp16(aG0 + kN, &As[nxt][soff0]);
      async_cp16(aG1 + kN, &As[nxt][soff1]);
      async_cp16(bG0 + kN, &Bs[nxt][soff0]);
      async_cp16(bG1 + kN, &Bs[nxt][soff1]);
    }
#else
    float4 ra0, ra1, rb0, rb1;
    if (more){
      ra0 = *(const float4*)(aG0 + kN); ra1 = *(const float4*)(aG1 + kN);
      rb0 = *(const float4*)(bG0 + kN); rb1 = *(const float4*)(bG1 + kN);
    }
#endif

    // ---- compute chunk: 4 k-steps x 8 WMMA ----
    const float* as_ = As[cur];
    const float* bs_ = Bs[cur];
    #pragma unroll
    for (int kk = 0; kk < KC; kk += 4){
      v2f a0 = *(const v2f*)(as_ + aOff0 + kk);
      v2f a1 = *(const v2f*)(as_ + aOff1 + kk);
      #pragma unroll
      for (int j = 0; j < 4; j++){
        v2f b = *(const v2f*)(bs_ + bOff[j] + kk);
        acc[0][j] = __builtin_amdgcn_wmma_f32_16x16x4_f32(
                      false, a0, false, b, (short)0, acc[0][j], false, false);
        acc[1][j] = __builtin_amdgcn_wmma_f32_16x16x4_f32(
                      false, a1, false, b, (short)0, acc[1][j], false, false);
      }
    }

#if ASYNC_OK
    __builtin_amdgcn_s_wait_asynccnt(0);
#else
    if (more){
      *(float4*)&As[nxt][soff0] = ra0;  *(float4*)&As[nxt][soff1] = ra1;
      *(float4*)&Bs[nxt][soff0] = rb0;  *(float4*)&Bs[nxt][soff1] = rb1;
    }
#endif
    __syncthreads();
  }

  // ---- epilogue: C layout VGPR v -> row (+half*8+v), col = l16 ----
  #pragma unroll
  for (int g = 0; g < 2; g++){
    const int rbase = mBase + wm*32 + g*16 + half*8;
    #pragma unroll
    for (int j = 0; j < 4; j++){
      const int col = nBase + wn*64 + j*16 + l16;
      if (col < N){
        const float bv = HB ? bias[col] : 0.f;
        #pragma unroll
        for (int v = 0; v < 8; v++){
          float val = acc[g][j][v] + bv;
          if (ACT == 1) val = softplusf_(val);
          const size_t idx = (size_t)(rbase + v) * N + col;
          if (HR) val += Res[idx];
          if (NT) __builtin_nontemporal_store(val, &C[idx]);
          else    C[idx] = val;
        }
      }
    }
  }
}

// ---------------- depthwise causal conv (D_CONV=4) + SiLU ----------------
__global__ void conv_silu_kernel(const float* __restrict__ xr, const float* __restrict__ cw,
                                 const float* __restrict__ cb, float* __restrict__ out)
{
  const int idx = blockIdx.x*256 + threadIdx.x;   // over ROWS*DI
  const int d   = idx % DI;
  const int row = idx / DI;
  const int l   = row & 1023;
  float acc = cb[d];
  const float w0 = cw[d*4+0], w1 = cw[d*4+1], w2 = cw[d*4+2], w3 = cw[d*4+3];
  const float* base = xr + (size_t)row*(2*DI) + d;
  if (l >= 3) acc += w0 * base[-3*(2*DI)];
  if (l >= 2) acc += w1 * base[-2*(2*DI)];
  if (l >= 1) acc += w2 * base[-1*(2*DI)];
  acc += w3 * base[0];
  out[(size_t)row*DI + d] = acc * sigmoidf_(acc);
}

// ---------------- segmented selective scan ----------------
// Linear diagonal recurrence x_t = dA_t*x + dBu_t decomposed over L into SEG
// segments.  Phase A (FINAL=false): zero-init local scans -> per-segment final
// state S and decay product P.  Phase B: sequential 8-step prefix combine.
// Phase C (FINAL=true): local scans with true init states -> gated output.
template<bool FINAL>
__global__ void scan_seg_kernel(const float* __restrict__ u,     // ROWS x DI
                                const float* __restrict__ dlt,   // ROWS x DI
                                const float* __restrict__ A_log, // DI x 16
                                const float* __restrict__ xdbl,  // ROWS x 80 (B||C at 48)
                                const float* __restrict__ Dv,    // DI
                                const float* __restrict__ xr,    // ROWS x 3072 (res at +1536)
                                const float* __restrict__ segI,  // init states  (FINAL)
                                float* __restrict__ segS,        // final states (!FINAL)
                                float* __restrict__ segP,        // decay prods  (!FINAL)
                                float* __restrict__ y)           // ROWS x DI    (FINAL)
{
  __shared__ float sBC[32][32];
  const int tid = threadIdx.x;
  const int d   = blockIdx.x*256 + tid;
  const int s   = blockIdx.y;
  const int b   = blockIdx.z;

  float Ad[16];
  #pragma unroll
  for (int n = 0; n < 16; n++) Ad[n] = -__expf(A_log[d*16 + n]);
  const float dval = Dv[d];

  const size_t sidx = ((size_t)(b*SEG + s)*DI + d)*16;
  float st[16], pr[16];
  #pragma unroll
  for (int n = 0; n < 16; n++){
    st[n] = FINAL ? segI[sidx + n] : 0.f;
    pr[n] = 1.f;
  }

  const int r0 = b*1024 + s*SEGLEN;
  for (int t0 = 0; t0 < SEGLEN; t0 += 32){
    __syncthreads();
    for (int i = tid; i < 32*32; i += 256){
      const int tl = i >> 5, j = i & 31;
      sBC[tl][j] = xdbl[(size_t)(r0 + t0 + tl)*80 + 48 + j];
    }
    __syncthreads();
    for (int tl = 0; tl < 32; tl++){
      const int row = r0 + t0 + tl;
      const float de = dlt[(size_t)row*DI + d];
      const float uv = u  [(size_t)row*DI + d];
      const float du = de*uv;
      float yv = 0.f;
      #pragma unroll
      for (int n = 0; n < 16; n++){
        const float dA = __expf(de*Ad[n]);
        st[n] = dA*st[n] + du*sBC[tl][n];
        if (FINAL) yv += st[n]*sBC[tl][16 + n];
        else       pr[n] *= dA;
      }
      if (FINAL){
        yv += uv*dval;
        const float r = xr[(size_t)row*(2*DI) + DI + d];
        y[(size_t)row*DI + d] = yv * (r * sigmoidf_(r));
      }
    }
  }

  if (!FINAL){
    #pragma unroll
    for (int n = 0; n < 16; n++){ segS[sidx + n] = st[n]; segP[sidx + n] = pr[n]; }
  }
}

// Phase B: per (b,d,n) prefix over segments.  2*DI*16 = 49152 threads.
__global__ void scan_combine_kernel(const float* __restrict__ segS,
                                    const float* __restrict__ segP,
                                    float* __restrict__ segI)
{
  const int idx = blockIdx.x*256 + threadIdx.x;  // b*(DI*16) + d*16 + n
  const int b   = idx / (DI*16);
  const int rem = idx % (DI*16);
  float x = 0.f;
  #pragma unroll
  for (int s = 0; s < SEG; s++){
    const size_t k = ((size_t)(b*SEG + s)*DI*16) + rem;
    segI[k] = x;
    x = segP[k]*x + segS[k];
  }
}

extern "C" void kernel_launch(void* const* d_in, const int* in_sizes, int n_in,
                              void* d_out, int out_size, void* d_ws, size_t ws_size,
                              hipStream_t stream)
{
  (void)in_sizes; (void)n_in; (void)out_size; (void)ws_size;
  const int*   ids    = (const int*)  d_in[0];
  /* d_in[1] = task (ignored) */
  const float* emb    = (const float*)d_in[2];   // VOC x 768
  const float* in_w   = (const float*)d_in[3];   // 2 x 3072 x 768
  const float* conv_w = (const float*)d_in[4];   // 2 x 1536 x 1 x 4
  const float* conv_b = (const float*)d_in[5];   // 2 x 1536
  const float* xp_w   = (const float*)d_in[6];   // 2 x 80 x 1536
  const float* dt_w   = (const float*)d_in[7];   // 2 x 1536 x 48
  const float* dt_b   = (const float*)d_in[8];   // 2 x 1536
  const float* A_log  = (const float*)d_in[9];   // 2 x 1536 x 16
  const float* Dvv    = (const float*)d_in[10];  // 2 x 1536
  const float* out_w  = (const float*)d_in[11];  // 2 x 768 x 1536
  const float* norm_w = (const float*)d_in[12];  // 2 x 768
  const float* normf  = (const float*)d_in[13];  // 768
  float* out = (float*)d_out;

  // workspace carve-up (~81 MB fp32)
  const size_t SEGSZ = (size_t)2*SEG*DI*16;      // 393216
  float* x    = (float*)d_ws;            // ROWS x 768
  float* xn   = x    + (size_t)ROWS*DM;
  float* xrr  = xn   + (size_t)ROWS*DM;  // ROWS x 3072 (xs || res)
  float* xsc  = xrr  + (size_t)ROWS*2*DI;// ROWS x 1536
  float* xdbl = xsc  + (size_t)ROWS*DI;  // ROWS x 80
  float* dl   = xdbl + (size_t)ROWS*80;  // ROWS x 1536
  float* yg   = dl   + (size_t)ROWS*DI;  // ROWS x 1536
  float* segS = yg   + (size_t)ROWS*DI;  // 2*SEG*DI*16
  float* segP = segS + SEGSZ;
  float* segI = segP + SEGSZ;

  embed_kernel<<<ROWS, 256, 0, stream>>>(ids, emb, x);

  for (int L = 0; L < 2; L++){
    rmsnorm_kernel<<<ROWS, 256, 0, stream>>>(x, norm_w + L*DM, xn);

    // in_proj: [2048x768] @ [3072x768]^T -> 2048x3072
    gemm_tiled<0,false,false,false><<<dim3(ROWS/128, (2*DI)/128), 256, 0, stream>>>(
        xn, DM, in_w + (size_t)L*2*DI*DM, DM, nullptr, nullptr, xrr, ROWS, 2*DI, DM);

    conv_silu_kernel<<<(ROWS*DI)/256, 256, 0, stream>>>(
        xrr, conv_w + L*DI*4, conv_b + L*DI, xsc);

    // x_proj: [2048x1536] @ [80x1536]^T -> 2048x80
    gemm_tiled<0,false,false,false><<<dim3(ROWS/128, 1), 256, 0, stream>>>(
        xsc, DI, xp_w + (size_t)L*80*DI, DI, nullptr, nullptr, xdbl, ROWS, 80, DI);

    // dt_proj + bias + softplus: [2048x48 view] @ [1536x48]^T -> 2048x1536
    gemm_tiled<1,true,false,false><<<dim3(ROWS/128, DI/128), 256, 0, stream>>>(
        xdbl, 80, dt_w + (size_t)L*DI*48, 48, dt_b + L*DI, nullptr, dl, ROWS, DI, 48);

    // segmented scan: A (local), B (combine), C (final + gating)
    scan_seg_kernel<false><<<dim3(DI/256, SEG, 2), 256, 0, stream>>>(
        xsc, dl, A_log + (size_t)L*DI*16, xdbl, Dvv + L*DI, xrr,
        nullptr, segS, segP, nullptr);
    scan_combine_kernel<<<(2*DI*16)/256, 256, 0, stream>>>(segS, segP, segI);
    scan_seg_kernel<true><<<dim3(DI/256, SEG, 2), 256, 0, stream>>>(
        xsc, dl, A_log + (size_t)L*DI*16, xdbl, Dvv + L*DI, xrr,
        segI, nullptr, nullptr, yg);

    // out_proj + residual: [2048x1536] @ [768x1536]^T + x -> x
    gemm_tiled<0,false,true,false><<<dim3(ROWS/128, DM/128), 256, 0, stream>>>(
        yg, DI, out_w + (size_t)L*DM*DI, DI, nullptr, x, x, ROWS, DM, DI);
  }

  rmsnorm_kernel<<<ROWS, 256, 0, stream>>>(x, normf, xn);

  // logits: [2048x768] @ [50264x768]^T -> 2048x50264, streaming (NT) stores
  gemm_tiled<0,false,false,true><<<dim3(ROWS/128, (VOC + 127)/128), 256, 0, stream>>>(
      xn, DM, emb, DM, nullptr, nullptr, out, ROWS, VOC, DM);
}